// NodeNetwork_61168924230358
// MI455X (gfx1250) — compile-verified
//
#include <hip/hip_runtime.h>
#include <cstdint>

#define B_   4
#define N_   20000
#define E_   320000
#define D_   64
#define H_   128
#define OUT_ 64
#define K1_  192   // 3*D concat input width
#define EPS_ 1e-5f

typedef __attribute__((ext_vector_type(2))) float v2f;
typedef __attribute__((ext_vector_type(8))) float v8f;

// ---------------------------------------------------------------------------
// Phase 1: edge gather + scatter-add.  One wave (32 lanes) per edge; each lane
// handles 2 consecutive features (D=64).  agg_in[dst] += w*nodes[src],
// agg_out[src] += w*nodes[dst], via GLOBAL_ATOMIC_ADD_F32 (L2-resident data).
// ---------------------------------------------------------------------------
__global__ void __launch_bounds__(256)
edge_scatter_kernel(const float* __restrict__ nodes,
                    const int*   __restrict__ edges,
                    const float* __restrict__ ew,
                    float*       __restrict__ agg_in,
                    float*       __restrict__ agg_out) {
    const int lane = threadIdx.x & 31;
    const int e    = blockIdx.x * 8 + (threadIdx.x >> 5);
    const int b    = blockIdx.y;
    if (e >= E_) return;
    const size_t eb  = (size_t)b * E_ + e;
    const int    src = edges[2 * eb + 0];
    const int    dst = edges[2 * eb + 1];
    const float  w   = ew[eb];
    const size_t sbase = ((size_t)b * N_ + src) * D_ + 2 * lane;
    const size_t dbase = ((size_t)b * N_ + dst) * D_ + 2 * lane;
    const float2 vs = *(const float2*)(nodes + sbase);
    const float2 vd = *(const float2*)(nodes + dbase);
    atomicAdd(agg_in  + dbase + 0, w * vs.x);
    atomicAdd(agg_in  + dbase + 1, w * vs.y);
    atomicAdd(agg_out + sbase + 0, w * vd.x);
    atomicAdd(agg_out + sbase + 1, w * vd.y);
}

// ---------------------------------------------------------------------------
// Phase 2: 4-layer MLP with f32 WMMA (16x16x4), layernorm + tanh per layer.
// One wave per 16-node tile.  xin/xout are LDS tiles [16 rows x dim].
// ---------------------------------------------------------------------------
__device__ __forceinline__ void layer_wmma(const float* __restrict__ W,
                                           const float* __restrict__ bias,
                                           const float* __restrict__ g,
                                           const float* __restrict__ t,
                                           const float* xin, int Kdim,
                                           float* xout, int Hdim, int lane) {
    const int m    = lane & 15;          // A row / B col / D col within tile
    const int koff = (lane >> 4) * 2;    // K sub-offset (ISA 16x4 f32 A layout)

    for (int c = 0; c < Hdim; c += 16) {
        v8f acc = {};
        for (int k = 0; k < Kdim; k += 4) {
            v2f a, bf;
            a.x  = xin[m * Kdim + k + koff];
            a.y  = xin[m * Kdim + k + koff + 1];
            bf.x = W[(size_t)(k + koff)     * Hdim + c + m];
            bf.y = W[(size_t)(k + koff + 1) * Hdim + c + m];
            // D = A(16x4) * B(4x16) + C ; exact f32 matrix op (CDNA5 VOP3P)
            acc = __builtin_amdgcn_wmma_f32_16x16x4_f32(
                false, a, false, bf, (short)0, acc, false, false);
        }
        // C/D layout: lanes 0-15 -> rows 0..7, lanes 16-31 -> rows 8..15
        const int rbase = (lane >> 4) * 8;
#pragma unroll
        for (int r = 0; r < 8; ++r)
            xout[(rbase + r) * Hdim + c + m] = acc[r];
    }
    __syncthreads();

    // LayerNorm(x + bias) * g + t, then tanh.  Two lanes per row; cross-half
    // reduction via wave32 shuffle-xor(16).
    const int row  = lane & 15;
    const int half = lane >> 4;
    const int j0   = half * (Hdim >> 1);
    const int j1   = j0 + (Hdim >> 1);
    float s = 0.f, s2 = 0.f;
    for (int j = j0; j < j1; ++j) {
        const float v = xout[row * Hdim + j] + bias[j];
        s += v; s2 += v * v;
    }
    s  += __shfl_xor(s, 16);
    s2 += __shfl_xor(s2, 16);
    const float mean = s / (float)Hdim;
    const float var  = s2 / (float)Hdim - mean * mean;
    const float rstd = rsqrtf(var + EPS_);
    for (int j = j0; j < j1; ++j) {
        float v = xout[row * Hdim + j] + bias[j];
        v = (v - mean) * rstd * g[j] + t[j];
        xout[row * Hdim + j] = tanhf(v);
    }
    __syncthreads();
}

__global__ void __launch_bounds__(32)
mlp_kernel(const float* __restrict__ agg_in, const float* __restrict__ agg_out,
           const float* __restrict__ nodes,
           const float* __restrict__ W1, const float* __restrict__ b1,
           const float* __restrict__ g1, const float* __restrict__ t1,
           const float* __restrict__ W2, const float* __restrict__ b2,
           const float* __restrict__ g2, const float* __restrict__ t2,
           const float* __restrict__ W3, const float* __restrict__ b3,
           const float* __restrict__ g3, const float* __restrict__ t3,
           const float* __restrict__ W4, const float* __restrict__ b4,
           const float* __restrict__ g4, const float* __restrict__ t4,
           float* __restrict__ out) {
    __shared__ float bufA[16 * K1_];   // 12 KB (also reused at width 128 / 64)
    __shared__ float bufB[16 * H_];    //  8 KB
    const int lane  = threadIdx.x;
    const int b     = blockIdx.y;
    const int node0 = blockIdx.x * 16;   // N % 16 == 0 -> EXEC stays all-1s

    // Build x = [agg_in | agg_out | nodes] for the 16-node tile.
    for (int i = lane; i < 16 * D_; i += 32) {
        const int    r    = i >> 6, j = i & 63;
        const size_t base = ((size_t)b * N_ + node0 + r) * D_ + j;
        bufA[r * K1_ + j]            = agg_in[base];
        bufA[r * K1_ + D_ + j]       = agg_out[base];
        bufA[r * K1_ + 2 * D_ + j]   = nodes[base];
    }
    __syncthreads();

    layer_wmma(W1, b1, g1, t1, bufA, K1_, bufB, H_,   lane);
    layer_wmma(W2, b2, g2, t2, bufB, H_,  bufA, H_,   lane);
    layer_wmma(W3, b3, g3, t3, bufA, H_,  bufB, H_,   lane);
    layer_wmma(W4, b4, g4, t4, bufB, H_,  bufA, OUT_, lane);

    for (int i = lane; i < 16 * OUT_; i += 32) {
        const int r = i >> 6, j = i & 63;
        out[((size_t)b * N_ + node0 + r) * OUT_ + j] = bufA[r * OUT_ + j];
    }
}

// ---------------------------------------------------------------------------
extern "C" void kernel_launch(void* const* d_in, const int* in_sizes, int n_in,
                              void* d_out, int out_size, void* d_ws, size_t ws_size,
                              hipStream_t stream) {
    (void)in_sizes; (void)n_in; (void)out_size; (void)ws_size;

    const float* nodes = (const float*)d_in[0];
    const int*   edges = (const int*)d_in[1];
    const float* ew    = (const float*)d_in[2];
    const float* W1 = (const float*)d_in[3],  *b1 = (const float*)d_in[4];
    const float* g1 = (const float*)d_in[5],  *t1 = (const float*)d_in[6];
    const float* W2 = (const float*)d_in[7],  *b2 = (const float*)d_in[8];
    const float* g2 = (const float*)d_in[9],  *t2 = (const float*)d_in[10];
    const float* W3 = (const float*)d_in[11], *b3 = (const float*)d_in[12];
    const float* g3 = (const float*)d_in[13], *t3 = (const float*)d_in[14];
    const float* W4 = (const float*)d_in[15], *b4 = (const float*)d_in[16];
    const float* g4 = (const float*)d_in[17], *t4 = (const float*)d_in[18];

    float* agg_in  = (float*)d_ws;
    float* agg_out = agg_in + (size_t)B_ * N_ * D_;
    const size_t agg_bytes = 2 * (size_t)B_ * N_ * D_ * sizeof(float);

    // Zero accumulators every call (atomics accumulate; must be deterministic).
    hipMemsetAsync(d_ws, 0, agg_bytes, stream);

    edge_scatter_kernel<<<dim3(E_ / 8, B_), 256, 0, stream>>>(
        nodes, edges, ew, agg_in, agg_out);

    mlp_kernel<<<dim3(N_ / 16, B_), 32, 0, stream>>>(
        agg_in, agg_out, nodes,
        W1, b1, g1, t1, W2, b2, g2, t2,
        W3, b3, g3, t3, W4, b4, g4, t4,
        (float*)d_out);
}